// GraphAttentionLayer_18949395710590
// MI455X (gfx1250) — compile-verified
//
#include <hip/hip_runtime.h>
#include <hip/hip_bf16.h>

#define BB 4
#define NN 1024
#define INF 256
#define OUTF 256
#define NH 8
#define HID 32
#define ALPHA 0.2f

typedef float v2f __attribute__((ext_vector_type(2)));
typedef float v8f __attribute__((ext_vector_type(8)));
typedef unsigned int u32x4 __attribute__((ext_vector_type(4)));
typedef int i32x8 __attribute__((ext_vector_type(8)));
typedef int i32x4 __attribute__((ext_vector_type(4)));

// ---------------------------------------------------------------------------
// Tensor Data Mover: issue a 2-D tile load (global -> LDS), data_size = 4B.
// d0flags carries data_size + LDS row-padding config (pad_interval/amount).
// Tensor dims are set huge (2^30) so no OOB clipping is ever triggered.
// Must be called with uniform (wave-invariant) arguments by ONE wave.
// Toolchain note: this clang-23 lane uses the 6-arg builtin
//   (u32x4 g0, i32x8 g1, i32x4 g2, i32x4 g3, i32x8 extra, i32 cpol)
// ---------------------------------------------------------------------------
__device__ __forceinline__ void tdm_load_2d(unsigned lds_off, const void* gaddr,
                                            int tile0, int tile1,
                                            long long stride0, unsigned d0flags) {
  const unsigned long long ga = (unsigned long long)gaddr;
  u32x4 g0;
  g0[0] = 1u;                                             // count=1 (valid)
  g0[1] = lds_off;                                        // lds_addr (bytes)
  g0[2] = (unsigned)(ga & 0xFFFFFFFFu);                   // global_addr lo
  g0[3] = (unsigned)((ga >> 32) & 0x01FFFFFFu) | (2u << 30); // addr hi | type=2
  i32x8 g1;
  g1[0] = (int)d0flags;                 // wg_mask=0 | data_size | pad cfg
  g1[1] = 0;                            // atomic_barrier=0 | tdim0[15:0]=0
  g1[2] = 0x4000;                       // tdim0 = 2^30 (hi) | tdim1 lo16 = 0
  g1[3] = 0x4000 | (tile0 << 16);       // tdim1 = 2^30 (hi) | tile_dim0
  g1[4] = tile1;                        // tile_dim1 | tile_dim2=0
  g1[5] = (int)(stride0 & 0xFFFFFFFF);  // tensor_dim0_stride lo32
  g1[6] = (int)(stride0 >> 32);         // stride0 hi16 | stride1 lo16 = 0
  g1[7] = 0;
  i32x4 z4 = {0, 0, 0, 0};
  i32x8 z8 = {0, 0, 0, 0, 0, 0, 0, 0};
  __builtin_amdgcn_tensor_load_to_lds(g0, g1, z4, z4, z8, 0);
}

#define TDM_F32_NOPAD          (2u << 16)
#define TDM_F32_PAD(ivl, amt)  ((2u << 16) | (1u << 20) | ((ivl) << 22) | ((amt) << 25))

// ---------------------------------------------------------------------------
// Kernel 1: g = h @ W   (per batch: [1024x256] @ [256x256], fp32 WMMA)
// Block: 256 threads = 8 waves. Block tile: 64 (M) x 256 (N).
// Tiles staged by the Tensor Data Mover (wave 0 issues, all waves consume).
// ---------------------------------------------------------------------------
__global__ __launch_bounds__(256) void k_gemm_g(const float* __restrict__ hin,
                                                const float* __restrict__ W,
                                                float* __restrict__ g) {
  __shared__ float As[64][68];    // 64 data + 4 pad dwords/row (TDM pad cfg)
  __shared__ float Bs[64][256];

  const int tid  = threadIdx.x;
  const int b    = blockIdx.x >> 4;           // 16 M-tiles per batch
  const int i0   = (blockIdx.x & 15) << 6;
  const int wave = tid >> 5, lane = tid & 31;
  const int iw   = wave & 3, nw = wave >> 2;
  const int m    = lane & 15, half = lane >> 4;

  const unsigned asOff = (unsigned)(size_t)(void*)&As[0][0];
  const unsigned bsOff = (unsigned)(size_t)(void*)&Bs[0][0];

  const v8f vzero = {0.f,0.f,0.f,0.f,0.f,0.f,0.f,0.f};
  v8f acc[8];
#pragma unroll
  for (int i = 0; i < 8; ++i) acc[i] = vzero;

  const float* hA = hin + ((size_t)b * NN + i0) * INF;

  for (int kt = 0; kt < 4; ++kt) {
    const int k0 = kt * 64;
    __syncthreads();
    if (wave == 0) {
      // A tile: h[b, i0:i0+64, k0:k0+64], row stride 256, pad 64->68 dwords
      tdm_load_2d(asOff, hA + k0, /*tile0=*/64, /*tile1=*/64,
                  /*stride0=*/INF, TDM_F32_PAD(5u /*64 dw*/, 3u /*4 dw*/));
      // B tile: W[k0:k0+64, 0:256], contiguous rows
      tdm_load_2d(bsOff, W + (size_t)k0 * OUTF, /*tile0=*/256, /*tile1=*/64,
                  /*stride0=*/OUTF, TDM_F32_NOPAD);
      __builtin_amdgcn_s_wait_tensorcnt(0);
    }
    __syncthreads();

#pragma unroll 4
    for (int k = 0; k < 64; k += 4) {
      // A frag 16x4: lanes 0-15 -> K=0,1 ; lanes 16-31 -> K=2,3
      v2f a;
      a.x = As[iw * 16 + m][k + 2 * half];
      a.y = As[iw * 16 + m][k + 2 * half + 1];
#pragma unroll
      for (int nf = 0; nf < 8; ++nf) {
        const int col = nw * 128 + nf * 16 + m;
        v2f bb;
        bb.x = Bs[k + 2 * half][col];
        bb.y = Bs[k + 2 * half + 1][col];
        acc[nf] = __builtin_amdgcn_wmma_f32_16x16x4_f32(
            false, a, false, bb, (short)0, acc[nf], false, false);
      }
    }
  }

  // D layout: VGPR r -> M=r (lanes 0-15) / M=r+8 (lanes 16-31), N=lane&15
#pragma unroll
  for (int nf = 0; nf < 8; ++nf) {
#pragma unroll
    for (int r = 0; r < 8; ++r) {
      const int row = i0 + iw * 16 + r + 8 * half;
      const int col = nw * 128 + nf * 16 + m;
      g[((size_t)b * NN + row) * OUTF + col] = acc[nf][r];
    }
  }
}

// ---------------------------------------------------------------------------
// Kernel 2: e_i[b,n,h] = <g[b,n,h,:], aw[0:32]>, e_j = <g, aw[32:64]>
// ---------------------------------------------------------------------------
__global__ __launch_bounds__(256) void k_ei_ej(const float* __restrict__ g,
                                               const float* __restrict__ aw,
                                               float* __restrict__ ei,
                                               float* __restrict__ ej) {
  const int t = blockIdx.x * 256 + threadIdx.x;      // (b*N+n)*NH + h
  const float* gp = g + (size_t)(t / NH) * OUTF + (t % NH) * HID;
  float si = 0.f, sj = 0.f;
#pragma unroll
  for (int d = 0; d < HID; d += 4) {
    float4 v  = *(const float4*)(gp + d);
    float4 wi = *(const float4*)(aw + d);
    float4 wj = *(const float4*)(aw + HID + d);
    si += v.x * wi.x + v.y * wi.y + v.z * wi.z + v.w * wi.w;
    sj += v.x * wj.x + v.y * wj.y + v.z * wj.z + v.w * wj.w;
  }
  ei[t] = si;
  ej[t] = sj;
}

// ---------------------------------------------------------------------------
// Kernel 3: column softmax stats (softmax over axis i, per (b,j,h)).
// Online single-pass max/sum; one thread per j; coalesced adj column scan.
// ---------------------------------------------------------------------------
__global__ __launch_bounds__(256) void k_colsoftmax(const float* __restrict__ adj,
                                                    const float* __restrict__ ei,
                                                    const float* __restrict__ ej,
                                                    float* __restrict__ cmax,
                                                    float* __restrict__ crecip) {
  __shared__ float eis[NN * NH];                     // 32 KB: e_i for batch b
  const int tid = threadIdx.x;
  const int b   = blockIdx.x >> 2;
  const int j   = ((blockIdx.x & 3) << 8) + tid;

  const float* eib = ei + (size_t)b * NN * NH;
#pragma unroll
  for (int p = 0; p < 8; ++p) {
    int idx = (tid + p * 256) << 2;
    *(float4*)&eis[idx] = *(const float4*)(eib + idx);
  }
  __syncthreads();

  float ejr[NH], mx[NH], sm[NH];
#pragma unroll
  for (int hh = 0; hh < NH; ++hh) {
    ejr[hh] = ej[((size_t)b * NN + j) * NH + hh];
    mx[hh]  = -__builtin_inff();
    sm[hh]  = 0.f;
  }

  const float* adjb = adj + (size_t)b * NN * NN + j;
  for (int i = 0; i < NN; ++i) {
    if (i + 16 < NN) __builtin_prefetch(adjb + (size_t)(i + 16) * NN, 0, 1);
    const float a = adjb[(size_t)i * NN];
    if (a != 0.f) {
#pragma unroll
      for (int hh = 0; hh < NH; ++hh) {
        float x = eis[i * NH + hh] + ejr[hh];
        x = (x >= 0.f) ? x : ALPHA * x;
        if (x > mx[hh]) {
          sm[hh] = sm[hh] * __expf(mx[hh] - x) + 1.f;
          mx[hh] = x;
        } else {
          sm[hh] += __expf(x - mx[hh]);
        }
      }
    }
  }
  const size_t o = ((size_t)b * NN + j) * NH;
#pragma unroll
  for (int hh = 0; hh < NH; ++hh) {
    cmax[o + hh]   = mx[hh];
    crecip[o + hh] = 1.f / sm[hh];
  }
}

// ---------------------------------------------------------------------------
// Kernel 4: out[b,i,h,:] = sum_j a[b,i,j,h] * g[b,j,h,:]   (fused, WMMA)
// adj tile staged by TDM (with hardware row padding 128->132 dwords);
// g tile pre-scaled by 1/sum cooperatively while the TDM is in flight.
// ---------------------------------------------------------------------------
__global__ __launch_bounds__(256) void k_aggregate(const float* __restrict__ adj,
                                                   const float* __restrict__ g,
                                                   const float* __restrict__ ei,
                                                   const float* __restrict__ ej,
                                                   const float* __restrict__ cmax,
                                                   const float* __restrict__ crecip,
                                                   float* __restrict__ out) {
  __shared__ float adjS[128][132];   // 128 data + 4 pad dwords/row (TDM pad)
  __shared__ float gS[128][32];      // pre-scaled by 1/sum
  __shared__ float cmS[128];
  __shared__ float cejS[128];

  const int tid  = threadIdx.x;
  const int b    = blockIdx.x >> 6;          // NH * 8 tiles per batch
  const int hh   = (blockIdx.x >> 3) & 7;
  const int i0   = (blockIdx.x & 7) << 7;
  const int wave = tid >> 5, lane = tid & 31;
  const int m    = lane & 15, half = lane >> 4;

  const unsigned adjOff = (unsigned)(size_t)(void*)&adjS[0][0];

  const int irow  = i0 + wave * 16 + m;      // A-matrix row for this lane
  const float eir = ei[((size_t)b * NN + irow) * NH + hh];

  const v8f vzero = {0.f,0.f,0.f,0.f,0.f,0.f,0.f,0.f};
  v8f acc0 = vzero, acc1 = vzero;

  for (int j0 = 0; j0 < NN; j0 += 128) {
    __syncthreads();
    // TDM: adj[b, i0:i0+128, j0:j0+128], row stride 1024, pad 128->132 dwords
    if (wave == 0) {
      tdm_load_2d(adjOff, adj + ((size_t)b * NN + i0) * NN + j0,
                  /*tile0=*/128, /*tile1=*/128, /*stride0=*/NN,
                  TDM_F32_PAD(6u /*128 dw*/, 3u /*4 dw*/));
    }
    // stage g[b, j0:j0+128, hh*32:+32], scaled by crecip (overlaps TDM)
#pragma unroll
    for (int p = 0; p < 4; ++p) {
      int idx = tid + p * 256;
      int r = idx >> 3, c4 = (idx & 7) << 2;
      const float rc = crecip[((size_t)b * NN + j0 + r) * NH + hh];
      float4 v = *(const float4*)(g + ((size_t)b * NN + j0 + r) * OUTF + hh * HID + c4);
      v.x *= rc; v.y *= rc; v.z *= rc; v.w *= rc;
      *(float4*)&gS[r][c4] = v;
    }
    if (tid < 128) {
      cmS[tid]  = cmax[((size_t)b * NN + j0 + tid) * NH + hh];
      cejS[tid] = ej[((size_t)b * NN + j0 + tid) * NH + hh];
    }
    if (wave == 0) __builtin_amdgcn_s_wait_tensorcnt(0);
    __syncthreads();

#pragma unroll 2
    for (int jk = 0; jk < 128; jk += 4) {
      const int jl = jk + 2 * half;
      // build attention-weight A-fragment (2 elems per lane)
      const float a0 = adjS[wave * 16 + m][jl];
      const float a1 = adjS[wave * 16 + m][jl + 1];
      float x0 = eir + cejS[jl];
      float x1 = eir + cejS[jl + 1];
      x0 = (x0 >= 0.f) ? x0 : ALPHA * x0;
      x1 = (x1 >= 0.f) ? x1 : ALPHA * x1;
      v2f af;
      af.x = (a0 != 0.f) ? __expf(x0 - cmS[jl]) : 0.f;
      af.y = (a1 != 0.f) ? __expf(x1 - cmS[jl + 1]) : 0.f;
      v2f b0, b1;
      b0.x = gS[jl][m];       b0.y = gS[jl + 1][m];
      b1.x = gS[jl][16 + m];  b1.y = gS[jl + 1][16 + m];
      acc0 = __builtin_amdgcn_wmma_f32_16x16x4_f32(
          false, af, false, b0, (short)0, acc0, false, false);
      acc1 = __builtin_amdgcn_wmma_f32_16x16x4_f32(
          false, af, false, b1, (short)0, acc1, false, false);
    }
  }

#pragma unroll
  for (int r = 0; r < 8; ++r) {
    const int row = i0 + wave * 16 + r + 8 * half;
    const size_t o = ((size_t)b * NN + row) * OUTF + hh * HID;
    out[o + m]      = acc0[r];
    out[o + 16 + m] = acc1[r];
  }
}

// ---------------------------------------------------------------------------
extern "C" void kernel_launch(void* const* d_in, const int* in_sizes, int n_in,
                              void* d_out, int out_size, void* d_ws, size_t ws_size,
                              hipStream_t stream) {
  const float* h_in = (const float*)d_in[0];   // [B, N, 256]
  const float* adj  = (const float*)d_in[1];   // [B, N, N, 1]
  const float* W    = (const float*)d_in[2];   // [256, 256]
  const float* aw   = (const float*)d_in[3];   // [64]
  float* out = (float*)d_out;                  // [B, N, 256]

  float* g      = (float*)d_ws;                       // B*N*256
  float* ei     = g + (size_t)BB * NN * OUTF;         // B*N*8
  float* ej     = ei + (size_t)BB * NN * NH;
  float* cmax   = ej + (size_t)BB * NN * NH;
  float* crecip = cmax + (size_t)BB * NN * NH;

  k_gemm_g<<<BB * (NN / 64), 256, 0, stream>>>(h_in, W, g);
  k_ei_ej<<<(BB * NN * NH) / 256, 256, 0, stream>>>(g, aw, ei, ej);
  k_colsoftmax<<<BB * (NN / 256), 256, 0, stream>>>(adj, ei, ej, cmax, crecip);
  k_aggregate<<<BB * NH * (NN / 128), 256, 0, stream>>>(adj, g, ei, ej, cmax,
                                                        crecip, out);
}